// TabM_with_PLE_31207232373401
// MI455X (gfx1250) — compile-verified
//
#include <hip/hip_runtime.h>
#include <hip/hip_bf16.h>

#define BATCH   2048
#define D_FEAT  100
#define NBINS   10
#define D_IN    1000
#define D_HID   512
#define KENS    32

typedef __attribute__((ext_vector_type(2))) float v2f;
typedef __attribute__((ext_vector_type(8))) float v8f;

#define SA_STRIDE 36   // 128 rows x 32 kk, padded: 36*4B = 144 ≡ 0 (mod 16)
#define SW_STRIDE 68   // 32 kk x 64 col,  padded: 68*4B = 272 ≡ 0 (mod 16)

// ---------------------------------------------------------------------------
// Kernel A: per-feature quantile bins (bitonic sort in LDS, linear interp).
// bins layout: [NBINS][D_FEAT] (matches jnp bins (T, d)).
// ---------------------------------------------------------------------------
__global__ __launch_bounds__(256)
void bins_kernel(const float* __restrict__ x, float* __restrict__ bins) {
    const int f = blockIdx.x;
    __shared__ float s[BATCH];
    for (int i = threadIdx.x; i < BATCH; i += 256)
        s[i] = x[(size_t)i * D_FEAT + f];
    __syncthreads();

    for (unsigned k = 2; k <= BATCH; k <<= 1) {
        for (unsigned j = k >> 1; j > 0; j >>= 1) {
            for (unsigned i = threadIdx.x; i < BATCH; i += 256) {
                unsigned ixj = i ^ j;
                if (ixj > i) {
                    float a = s[i], b = s[ixj];
                    bool up = ((i & k) == 0);
                    if (up ? (a > b) : (a < b)) { s[i] = b; s[ixj] = a; }
                }
            }
            __syncthreads();
        }
    }

    if (threadIdx.x < NBINS) {
        int t = threadIdx.x;
        float pos = (float)t * ((float)(BATCH - 1) / (float)(NBINS - 1));
        int i0 = (int)floorf(pos);
        if (i0 > BATCH - 2) i0 = BATCH - 2;
        float frac = pos - (float)i0;
        bins[t * D_FEAT + f] = s[i0] + frac * (s[i0 + 1] - s[i0]);
    }
}

// ---------------------------------------------------------------------------
// Kernel B: piecewise-linear encoding. enc: [BATCH][D_FEAT*NBINS], feat-major.
// ---------------------------------------------------------------------------
__global__ __launch_bounds__(256)
void ple_kernel(const float* __restrict__ x, const float* __restrict__ bins,
                float* __restrict__ enc) {
    int idx = blockIdx.x * 256 + threadIdx.x;           // b * D_FEAT + f
    if (idx >= BATCH * D_FEAT) return;
    int b = idx / D_FEAT, f = idx % D_FEAT;
    float v = x[idx];
    float* o = enc + (size_t)b * D_IN + f * NBINS;
    o[0] = 0.0f;
#pragma unroll
    for (int t = 1; t < NBINS; ++t) {
        float lo = bins[(t - 1) * D_FEAT + f];
        float hi = bins[t * D_FEAT + f];
        float val = (v >= hi && t < NBINS - 1) ? 1.0f : 0.0f;
        if (v >= lo && v < hi) val = (v - lo) / (hi - lo + 1e-9f);
        o[t] = val;
    }
}

// ---------------------------------------------------------------------------
// Kernel C: BatchEnsemble layer  Y[k] = relu( ((A_k ⊙ R[k]) @ W) * S[k] + Bb[k] )
// fp32 WMMA 16x16x4.  Block: 128(b) x 64(o) per k; 8 waves in a 4x2 grid,
// each wave owns 32x32 (4 accumulators -> each LDS fragment feeds 2 WMMAs).
// Double-buffered LDS staging, one barrier per 32-deep K chunk; next chunk's
// global loads issued before the current chunk's WMMAs so VMEM latency hides
// under the matrix pipe.
// ---------------------------------------------------------------------------
template<int DK, bool PER_K_A>
__global__ __launch_bounds__(256)
void be_gemm_relu(const float* __restrict__ A,    // PER_K_A ? [K][B][DK] : [B][DK]
                  const float* __restrict__ R,    // [K][DK]
                  const float* __restrict__ W,    // [DK][D_HID]
                  const float* __restrict__ S,    // [K][D_HID]
                  const float* __restrict__ Bb,   // [K][D_HID]
                  float* __restrict__ Y)          // [K][B][D_HID]
{
    const int k     = blockIdx.z;
    const int bBase = blockIdx.x * 128;
    const int oBase = blockIdx.y * 64;
    const int tid   = threadIdx.x;
    const int lane  = tid & 31;
    const int wave  = tid >> 5;
    const int bRow  = (wave & 3) * 32;      // wave's 32-row slab
    const int oCol  = (wave >> 2) * 32;     // wave's 32-col slab
    const int hl    = lane & 15;
    const int hi    = lane >> 4;            // 0: K pair {0,1}; 1: K pair {2,3}

    __shared__ float sA[2][128 * SA_STRIDE];
    __shared__ float sW[2][32 * SW_STRIDE];

    const float* Ak = A + (PER_K_A ? (size_t)k * (size_t)BATCH * DK : (size_t)0);
    const float* Rk = R + (size_t)k * DK;

    v8f c00 = {}, c01 = {}, c10 = {}, c11 = {};

    // ---- stage one 32-deep chunk (i0 = chunk base) into buffer `buf` ----
    auto stage = [&](int i0, int buf) {
        // A tile: 128 rows x 32 kk, pre-scaled by R[k]; float4 fast path.
        for (int t = tid; t < 128 * 8; t += 256) {
            int r = t >> 3, seg = (t & 7) * 4;
            int gi = i0 + seg;
            const float* src = &Ak[(size_t)(bBase + r) * DK + gi];
            float4 v = {0.f, 0.f, 0.f, 0.f};
            float4 rv = {0.f, 0.f, 0.f, 0.f};
            if (gi + 3 < DK) {
                v  = *(const float4*)src;
                rv = *(const float4*)&Rk[gi];
            } else {
                if (gi + 0 < DK) { v.x = src[0]; rv.x = Rk[gi + 0]; }
                if (gi + 1 < DK) { v.y = src[1]; rv.y = Rk[gi + 1]; }
                if (gi + 2 < DK) { v.z = src[2]; rv.z = Rk[gi + 2]; }
            }
            float4 o;
            o.x = v.x * rv.x; o.y = v.y * rv.y;
            o.z = v.z * rv.z; o.w = v.w * rv.w;
            *(float4*)&sA[buf][r * SA_STRIDE + seg] = o;
        }
        // W tile: 32 kk x 64 cols; cols always in range, guard rows only.
        for (int t = tid; t < 32 * 16; t += 256) {
            int kk = t >> 4, seg = (t & 15) * 4;
            int gi = i0 + kk;
            float4 v = {0.f, 0.f, 0.f, 0.f};
            if (gi < DK) v = *(const float4*)&W[(size_t)gi * D_HID + oBase + seg];
            *(float4*)&sW[buf][kk * SW_STRIDE + seg] = v;
        }
    };

    const int nch = (DK + 31) / 32;
    stage(0, 0);
    __syncthreads();

    for (int ch = 0; ch < nch; ++ch) {
        const int cur = ch & 1;
        if (ch + 1 < nch) stage((ch + 1) * 32, cur ^ 1);

        const float* sAb = sA[cur];
        const float* sWb = sW[cur];

#pragma unroll
        for (int kk0 = 0; kk0 < 32; kk0 += 4) {
            const int ak = kk0 + hi * 2;   // lanes 16-31 hold K+2,K+3 (ISA layout)
            v2f a0 = *(const v2f*)&sAb[(bRow + hl) * SA_STRIDE + ak];
            v2f a1 = *(const v2f*)&sAb[(bRow + 16 + hl) * SA_STRIDE + ak];
            v2f b0, b1;
            b0.x = sWb[ak * SW_STRIDE + oCol + hl];
            b0.y = sWb[(ak + 1) * SW_STRIDE + oCol + hl];
            b1.x = sWb[ak * SW_STRIDE + oCol + 16 + hl];
            b1.y = sWb[(ak + 1) * SW_STRIDE + oCol + 16 + hl];
            c00 = __builtin_amdgcn_wmma_f32_16x16x4_f32(
                      false, a0, false, b0, (short)0, c00, false, false);
            c01 = __builtin_amdgcn_wmma_f32_16x16x4_f32(
                      false, a0, false, b1, (short)0, c01, false, false);
            c10 = __builtin_amdgcn_wmma_f32_16x16x4_f32(
                      false, a1, false, b0, (short)0, c10, false, false);
            c11 = __builtin_amdgcn_wmma_f32_16x16x4_f32(
                      false, a1, false, b1, (short)0, c11, false, false);
        }
        __syncthreads();
    }

    // epilogue: *S[k] + Bb[k], relu, store.  C layout: VGPR v -> row v (lanes
    // 0-15) / row v+8 (lanes 16-31); col = lane%16.
    const float* Sk = S  + (size_t)k * D_HID;
    const float* Bk = Bb + (size_t)k * D_HID;
    float* Yk = Y + (size_t)k * BATCH * D_HID;

    const int col0 = oBase + oCol + hl;
    const int col1 = col0 + 16;
    const float s0 = Sk[col0], bb0 = Bk[col0];
    const float s1 = Sk[col1], bb1 = Bk[col1];

    const int rowA = bBase + bRow + hi * 8;        // sub-tile 0 rows
    const int rowB = rowA + 16;                    // sub-tile 1 rows
#pragma unroll
    for (int v = 0; v < 8; ++v) {
        float y;
        y = c00[v] * s0 + bb0; y = y > 0.f ? y : 0.f;
        Yk[(size_t)(rowA + v) * D_HID + col0] = y;
        y = c01[v] * s1 + bb1; y = y > 0.f ? y : 0.f;
        Yk[(size_t)(rowA + v) * D_HID + col1] = y;
        y = c10[v] * s0 + bb0; y = y > 0.f ? y : 0.f;
        Yk[(size_t)(rowB + v) * D_HID + col0] = y;
        y = c11[v] * s1 + bb1; y = y > 0.f ? y : 0.f;
        Yk[(size_t)(rowB + v) * D_HID + col1] = y;
    }
}

// ---------------------------------------------------------------------------
// Kernel D: head:  out[b] = mean_k( sum_d X2[k][b][d]*Wh[k][d] + bh[k] )
// One block per batch row; deterministic LDS tree reduction (no atomics).
// ---------------------------------------------------------------------------
__global__ __launch_bounds__(256)
void head_kernel(const float* __restrict__ X2, const float* __restrict__ Wh,
                 const float* __restrict__ bh, float* __restrict__ out) {
    const int b = blockIdx.x;
    float acc = 0.0f;
    for (int idx = threadIdx.x; idx < KENS * D_HID; idx += 256) {
        int k = idx >> 9, d = idx & (D_HID - 1);
        acc += X2[((size_t)k * BATCH + b) * D_HID + d] * Wh[k * D_HID + d];
    }
    __shared__ float red[256];
    red[threadIdx.x] = acc;
    __syncthreads();
    for (int s = 128; s > 0; s >>= 1) {
        if (threadIdx.x < s) red[threadIdx.x] += red[threadIdx.x + s];
        __syncthreads();
    }
    if (threadIdx.x == 0) {
        float bs = 0.0f;
        for (int k = 0; k < KENS; ++k) bs += bh[k];
        out[b] = (red[0] + bs) * (1.0f / (float)KENS);
    }
}

// ---------------------------------------------------------------------------
extern "C" void kernel_launch(void* const* d_in, const int* in_sizes, int n_in,
                              void* d_out, int out_size, void* d_ws, size_t ws_size,
                              hipStream_t stream) {
    (void)in_sizes; (void)n_in; (void)out_size; (void)ws_size;

    const float* x  = (const float*)d_in[0];
    const float* R1 = (const float*)d_in[1];
    const float* W1 = (const float*)d_in[2];
    const float* S1 = (const float*)d_in[3];
    const float* B1 = (const float*)d_in[4];
    const float* R2 = (const float*)d_in[5];
    const float* W2 = (const float*)d_in[6];
    const float* S2 = (const float*)d_in[7];
    const float* B2 = (const float*)d_in[8];
    const float* Wh = (const float*)d_in[9];
    const float* bh = (const float*)d_in[10];
    float* out = (float*)d_out;

    // workspace layout (256B aligned)
    char* ws = (char*)d_ws;
    float* bins = (float*)(ws);                                    //   4 KB
    float* enc  = (float*)(ws + 4096);                             //   8 MB
    float* X1   = (float*)(ws + 4096 + (size_t)BATCH * D_IN * 4);  // 128 MB
    float* X2   = X1 + (size_t)KENS * BATCH * D_HID;               // 128 MB

    bins_kernel<<<D_FEAT, 256, 0, stream>>>(x, bins);
    ple_kernel<<<(BATCH * D_FEAT + 255) / 256, 256, 0, stream>>>(x, bins, enc);

    dim3 g1(BATCH / 128, D_HID / 64, KENS);
    be_gemm_relu<D_IN,  false><<<g1, 256, 0, stream>>>(enc, R1, W1, S1, B1, X1);
    be_gemm_relu<D_HID, true ><<<g1, 256, 0, stream>>>(X1,  R2, W2, S2, B2, X2);

    head_kernel<<<BATCH, 256, 0, stream>>>(X2, Wh, bh, out);
}